// MedianPool2d_40329742909624
// MI455X (gfx1250) — compile-verified
//
#include <hip/hip_runtime.h>
#include <hip/hip_bf16.h>
#include <stdint.h>

// Problem constants (fixed by the reference): x is [16, 3, 512, 512] fp32,
// 3x3 median, stride 1, reflect pad 1 -> output same shape.
#define Hc 512
#define Wc 512
#define PLANES 48            // 16 * 3

// Tiling: 64 x 32 outputs per block, 256 threads, 8 outputs/thread (vertical).
#define TX 64
#define TY 32
#define RPT 8
#define LW (TX + 2)          // 66
#define LH (TY + 2)          // 34
#define NTHREADS 256

#if defined(__has_builtin)
#if __has_builtin(__builtin_amdgcn_global_load_async_to_lds_b32)
#define HAVE_ASYNC_LDS 1
#endif
#endif

typedef __attribute__((address_space(1))) int g_int_t;
typedef __attribute__((address_space(3))) int l_int_t;

__device__ __forceinline__ float med3f(float a, float b, float c) {
  return fmaxf(fminf(a, b), fminf(fmaxf(a, b), c));
}

struct Triple {
  float mn, md, mx;
};

// Full sort of 3 elements: 3 compare-exchanges = 6 v_min/v_max.
__device__ __forceinline__ Triple hsort3(const float* __restrict__ p) {
  float a = p[0], b = p[1], c = p[2];
  float t;
  t = fminf(a, b); b = fmaxf(a, b); a = t;
  t = fminf(b, c); c = fmaxf(b, c); b = t;
  t = fminf(a, b); b = fmaxf(a, b); a = t;
  Triple r;
  r.mn = a; r.md = b; r.mx = c;
  return r;
}

__global__ __launch_bounds__(NTHREADS)
void median3x3_kernel(const float* __restrict__ x, float* __restrict__ out) {
  __shared__ float tile[LW * LH];

  const int tx = threadIdx.x;             // 0..63
  const int ty = threadIdx.y;             // 0..3
  const int tid = ty * TX + tx;
  const int x0 = blockIdx.x * TX;
  const int y0 = blockIdx.y * TY;
  const size_t plane = (size_t)blockIdx.z * (size_t)(Hc * Wc);
  const float* __restrict__ src = x + plane;
  float* __restrict__ dst = out + plane;

  // gfx1250 prefetch (lowers to global_prefetch_b8).
  __builtin_prefetch(src + (size_t)((y0 > 0) ? (y0 - 1) : 0) * Wc + x0, 0, 1);

  // Stage the (TX+2) x (TY+2) halo tile into LDS with reflect-101 borders.
  for (int i = tid; i < LW * LH; i += NTHREADS) {
    const int r = i / LW;
    const int c = i - r * LW;
    int gy = y0 + r - 1;
    gy = (gy < 0) ? -gy : ((gy >= Hc) ? (2 * Hc - 2 - gy) : gy);
    int gx = x0 + c - 1;
    gx = (gx < 0) ? -gx : ((gx >= Wc) ? (2 * Wc - 2 - gx) : gx);
    const float* gp = src + (size_t)gy * Wc + gx;
#if defined(HAVE_ASYNC_LDS)
    // Async DMA global -> LDS, no VGPR round trip; tracked with ASYNCcnt.
    __builtin_amdgcn_global_load_async_to_lds_b32(
        (g_int_t*)gp,
        (l_int_t*)&tile[i],
        0, 0);
#else
    tile[i] = *gp;
#endif
  }
#if defined(HAVE_ASYNC_LDS)
#if __has_builtin(__builtin_amdgcn_s_wait_asynccnt)
  __builtin_amdgcn_s_wait_asynccnt(0);
#else
  asm volatile("s_wait_asynccnt 0" ::: "memory");
#endif
#endif
  __syncthreads();

  // Each thread: 8 vertical outputs at column tx. Roll sorted row-triples in
  // registers: 10 row sorts + 8 combines per thread (Smith 3x3 median net).
  const int jbase = ty * RPT;             // first local output row
  const float* col = &tile[tx];           // tile col tx == global col x0+tx-1

  Triple s0 = hsort3(&col[(jbase + 0) * LW]);
  Triple s1 = hsort3(&col[(jbase + 1) * LW]);

  float* orow = dst + (size_t)(y0 + jbase) * Wc + (x0 + tx);
#pragma unroll
  for (int k = 0; k < RPT; ++k) {
    Triple s2 = hsort3(&col[(jbase + k + 2) * LW]);
    const float lo = fmaxf(fmaxf(s0.mn, s1.mn), s2.mn);
    const float hi = fminf(fminf(s0.mx, s1.mx), s2.mx);
    const float md = med3f(s0.md, s1.md, s2.md);
    // Output is write-once: non-temporal store keeps the 192MB L2 for the
    // (re-used) input halos instead of write data.
    __builtin_nontemporal_store(med3f(lo, md, hi), &orow[(size_t)k * Wc]);
    s0 = s1;
    s1 = s2;
  }
}

extern "C" void kernel_launch(void* const* d_in, const int* in_sizes, int n_in,
                              void* d_out, int out_size, void* d_ws, size_t ws_size,
                              hipStream_t stream) {
  (void)in_sizes; (void)n_in; (void)out_size; (void)d_ws; (void)ws_size;
  const float* xin = (const float*)d_in[0];
  float* out = (float*)d_out;

  dim3 block(TX, TY / RPT, 1);                 // (64, 4) = 256 threads = 8 waves
  dim3 grid(Wc / TX, Hc / TY, PLANES);         // (8, 16, 48)
  hipLaunchKernelGGL(median3x3_kernel, grid, block, 0, stream, xin, out);
}